// GetValueV2_22445499089567
// MI455X (gfx1250) — compile-verified
//
#include <hip/hip_runtime.h>

// GetValueV2: deformable bilinear sampling.
//   x      : (8, 64, 128, 128) f32
//   offset : (8, 18, 128, 128) f32   (first 9 ch = dy, last 9 ch = dx)
//   out    : (8, 64, 128, 128, 9) f32
// Memory-bound: 302 MB NT streaming stores + L2-resident gathers of x.
//   - per block: one (b, h, 32-wide w tile)
//   - Phase 0: TDM tensor_load_to_lds of the 18x32 strided offset tile
//   - Phase 1: 288 points' bilinear taps/weights once into packed LDS
//   - Phase 2: point x 4-channel quads; 2x ds_load_b128 metadata, 16 gathers,
//              4 coalesced non-temporal stores; prefetch next quad's plane

typedef unsigned int v4u __attribute__((ext_vector_type(4)));
typedef int          v4i __attribute__((ext_vector_type(4)));
typedef int          v8i __attribute__((ext_vector_type(8)));

#define H_     128
#define W_     128
#define C_     64
#define NPTS   9
#define NOFFCH 18
#define TILE_W 32
#define BLOCK  256
#define PTS    (TILE_W * NPTS)   // 288 points per tile
#define PLANE  (H_ * W_)         // 16384
#define OPLANE (PLANE * NPTS)    // 147456 floats per (b,c) output plane

__global__ __launch_bounds__(BLOCK) void getvaluev2_kernel(
    const float* __restrict__ x,
    const float* __restrict__ off,
    float* __restrict__ out)
{
    __shared__ float  sh_off[NOFFCH * TILE_W];  // staged offset tile [ch][j]
    __shared__ int4   sh_idx[PTS];              // 4 clamped flat tap indices
    __shared__ float4 sh_wt[PTS];               // 4 weights (0 if tap invalid)

    const int wt  = blockIdx.x;            // w tile: 0..3
    const int h   = blockIdx.y;            // 0..127
    const int b   = blockIdx.z;            // batch
    const int w0  = wt * TILE_W;
    const int tid = threadIdx.x;

    // ---------------- Phase 0: stage offset tile into LDS ----------------
    if (tid < 32) {  // wave 0 only (uniform branch under wave32)
#if __has_builtin(__builtin_amdgcn_tensor_load_to_lds)
        const float* gsrc =
            off + ((size_t)b * NOFFCH) * PLANE + (size_t)h * W_ + w0;
        const uint64_t ga  = (uint64_t)(uintptr_t)gsrc;
        const uint32_t lds = (uint32_t)(uintptr_t)(&sh_off[0]); // LDS byte offs

        v4u g0;
        g0[0] = 1u;                                   // count=1, user mode
        g0[1] = lds;                                  // lds_addr
        g0[2] = (uint32_t)(ga & 0xFFFFFFFFu);         // global_addr[31:0]
        g0[3] = (uint32_t)((ga >> 32) & 0x01FFFFFFu)  // global_addr[56:32]
              | 0x80000000u;                          // type = 2 ("image")

        v8i g1;
        g1[0] = (int)(2u << 16);                      // data_size = 4 bytes
        g1[1] = (int)((uint32_t)TILE_W << 16);        // tensor_dim0 lo16 = 32
        g1[2] = (int)((uint32_t)NOFFCH << 16);        // dim0 hi | tensor_dim1 lo16 = 18
        g1[3] = (int)((uint32_t)TILE_W << 16);        // dim1 hi | tile_dim0 = 32
        g1[4] = NOFFCH;                               // tile_dim1 = 18 | tile_dim2 = 0
        g1[5] = PLANE;                                // tensor_dim0_stride = 16384 elems
        g1[6] = 0;
        g1[7] = 0;

        v4i gz = {0, 0, 0, 0};                        // groups 2/3 unused (2D)
  #if defined(__clang_major__) && (__clang_major__ >= 23)
        v8i g4z = {0, 0, 0, 0, 0, 0, 0, 0};
        __builtin_amdgcn_tensor_load_to_lds(g0, g1, gz, gz, g4z, 0);
  #else
        __builtin_amdgcn_tensor_load_to_lds(g0, g1, gz, gz, 0);
  #endif
        __builtin_amdgcn_s_wait_tensorcnt(0);
#else
        for (int i = tid; i < NOFFCH * TILE_W; i += 32) {
            const int ch = i / TILE_W, j = i % TILE_W;
            sh_off[i] = off[((size_t)b * NOFFCH + ch) * PLANE
                            + (size_t)h * W_ + (w0 + j)];
        }
#endif
    }
    __syncthreads();

    // ----- Phase 1: taps + weights per point (amortized over 64 channels) -----
    for (int p = tid; p < PTS; p += BLOCK) {
        const int j = p / NPTS;
        const int n = p - j * NPTS;
        const float offy = sh_off[n * TILE_W + j];
        const float offx = sh_off[(n + NPTS) * TILE_W + j];
        // base grid (STRIDE=1, 1-indexed), clip, normalize, unnormalize
        float py = (float)(h + 1) + offy;
        float px = (float)(w0 + j + 1) + offx;
        py = fminf(fmaxf(py, 0.0f), (float)(H_ - 1));
        px = fminf(fmaxf(px, 0.0f), (float)(W_ - 1));
        const float gy = py * (2.0f / (float)(H_ - 1)) - 1.0f;
        const float gx = px * (2.0f / (float)(W_ - 1)) - 1.0f;
        const float iy = ((gy + 1.0f) * (float)H_ - 1.0f) * 0.5f;
        const float ix = ((gx + 1.0f) * (float)W_ - 1.0f) * 0.5f;
        const float y0f = floorf(iy), x0f = floorf(ix);
        const float wy = iy - y0f, wx = ix - x0f;
        const int y0 = (int)y0f, x0i = (int)x0f;
        const int y1 = y0 + 1,  x1i = x0i + 1;
        const bool vy0 = (y0  >= 0) && (y0  < H_);
        const bool vy1 = (y1  >= 0) && (y1  < H_);
        const bool vx0 = (x0i >= 0) && (x0i < W_);
        const bool vx1 = (x1i >= 0) && (x1i < W_);
        const int cy0 = min(max(y0, 0),  H_ - 1);
        const int cy1 = min(max(y1, 0),  H_ - 1);
        const int cx0 = min(max(x0i, 0), W_ - 1);
        const int cx1 = min(max(x1i, 0), W_ - 1);
        sh_idx[p] = make_int4(cy0 * W_ + cx0, cy0 * W_ + cx1,
                              cy1 * W_ + cx0, cy1 * W_ + cx1);
        const float omwx = 1.0f - wx, omwy = 1.0f - wy;
        sh_wt[p] = make_float4((vy0 && vx0) ? omwx * omwy : 0.0f,
                               (vy0 && vx1) ? wx   * omwy : 0.0f,
                               (vy1 && vx0) ? omwx * wy   : 0.0f,
                               (vy1 && vx1) ? wx   * wy   : 0.0f);
    }
    __syncthreads();

    // ----- Phase 2: per point, 4 channel planes per iteration -----
    const float* xb = x + ((size_t)b * C_) * PLANE;
    const size_t obase =
        (((size_t)b * C_) * (size_t)PLANE + (size_t)h * W_ + w0) * (size_t)NPTS;

    int cq = 0;                    // channel quad 0..15
    int p  = tid;                  // BLOCK(256) < PTS(288): starts in cq=0
    const int totalq = (C_ / 4) * PTS;    // 4608 work items per block
    for (int e = tid; e < totalq; e += BLOCK) {
        const int4   i4 = sh_idx[p];      // one ds_load_b128
        const float4 w4 = sh_wt[p];       // one ds_load_b128
        const float* xp = xb + ((size_t)cq << 16);          // cq*4 planes in
        float*       op = out + obase + (size_t)cq * (4u * OPLANE) + (size_t)p;
        if (cq < (C_ / 4) - 1)  // warm next quad's first plane (L2 -> L0)
            __builtin_prefetch(xp + 4 * PLANE + i4.x, 0, 0);
#pragma unroll
        for (int k = 0; k < 4; ++k) {
            float r = xp[i4.x] * w4.x;
            r = fmaf(xp[i4.y], w4.y, r);
            r = fmaf(xp[i4.z], w4.z, r);
            r = fmaf(xp[i4.w], w4.w, r);
            // consecutive lanes -> consecutive p -> contiguous 128B stores;
            // NT hint: streamed output must not evict L2-resident x
            __builtin_nontemporal_store(r, op);
            xp += PLANE;
            op += OPLANE;
        }
        p += BLOCK;
        if (p >= PTS) { p -= PTS; ++cq; }
    }
}

extern "C" void kernel_launch(void* const* d_in, const int* in_sizes, int n_in,
                              void* d_out, int out_size, void* d_ws, size_t ws_size,
                              hipStream_t stream) {
    (void)n_in; (void)out_size; (void)d_ws; (void)ws_size;
    const float* x   = (const float*)d_in[0];
    const float* off = (const float*)d_in[1];
    float* out       = (float*)d_out;
    const int B = in_sizes[0] / (C_ * H_ * W_);   // 8
    dim3 grid(W_ / TILE_W, H_, B);                // 4 x 128 x 8 = 4096 blocks
    getvaluev2_kernel<<<grid, BLOCK, 0, stream>>>(x, off, out);
}